// Interpolator_678604832984
// MI455X (gfx1250) — compile-verified
//
#include <hip/hip_runtime.h>
#include <stdint.h>

// Interpolator: out[b,t,:] = lerp(x[b,idx,:], x[b,idx+1,:], w(t))
// B=8, T_IN=4096, D=512 fp32; out_len = 8191. Bandwidth-bound
// (~200 MB HBM traffic -> ~9us at 23.3 TB/s).
//
// CDNA5 structure:
//  - one block per output time t: binary search done once (scalarized,
//    SALU + s_load), amortized over all B batches
//  - async global->LDS staging (global_load_async_to_lds_b128, ASYNCcnt)
//    double-buffered across the batch loop: issue b+1, s_wait_asynccnt 2,
//    consume b  (async loads complete in order per wave)
//  - non-temporal B128 output stores keep x resident in L2

#define D_DIM   512
#define THREADS 128   // one float4 per thread covers D=512

typedef float v4f __attribute__((ext_vector_type(4)));

__device__ __forceinline__ void async_ld16(uint32_t lds_off, const void* gaddr) {
    asm volatile("global_load_async_to_lds_b128 %0, %1, off"
                 :: "v"(lds_off), "v"((uint64_t)(uintptr_t)gaddr) : "memory");
}

__device__ __forceinline__ void wait_async_le2() {
    asm volatile("s_wait_asynccnt 2" ::: "memory");
}
__device__ __forceinline__ void wait_async_le0() {
    asm volatile("s_wait_asynccnt 0" ::: "memory");
}

__global__ __launch_bounds__(THREADS)
void interp_async_kernel(const float* __restrict__ x,
                         const int*   __restrict__ tl,
                         float*       __restrict__ out,
                         int T_in, int out_len, int B)
{
    // two buffers, each holds left row [0..D) and right row [D..2D)
    __shared__ float sbuf[2][2 * D_DIM];

    const int t   = blockIdx.x;   // output time index
    const int tid = threadIdx.x;

    // searchsorted(tl, t, side='right') - 1, clipped to [0, T_in-2].
    // blockIdx-uniform -> compiler scalarizes (SALU + scalar loads).
    int lo = 0, hi = T_in;
    while (lo < hi) {
        int mid = (lo + hi) >> 1;
        if (tl[mid] <= t) lo = mid + 1; else hi = mid;
    }
    int idx = lo - 1;
    idx = idx < 0 ? 0 : idx;
    idx = idx > T_in - 2 ? T_in - 2 : idx;

    const float t0 = (float)tl[idx];
    const float t1 = (float)tl[idx + 1];
    const float w  = ((float)t - t0) / (t1 - t0);

    // Per-lane LDS byte offsets (generic LDS pointer keeps offset in addr[31:0]).
    const uint32_t ldsL[2] = { (uint32_t)(uintptr_t)(&sbuf[0][tid * 4]),
                               (uint32_t)(uintptr_t)(&sbuf[1][tid * 4]) };
    const uint32_t ldsR[2] = { (uint32_t)(uintptr_t)(&sbuf[0][D_DIM + tid * 4]),
                               (uint32_t)(uintptr_t)(&sbuf[1][D_DIM + tid * 4]) };

    // Rows idx and idx+1 are contiguous: 4 KB per batch.
    const int lane4 = tid * 4;

    // Prologue: stage batch 0 into buffer 0.
    {
        const float* base = x + ((size_t)idx) * D_DIM;   // b = 0
        async_ld16(ldsL[0], base + lane4);
        async_ld16(ldsR[0], base + D_DIM + lane4);
    }

    for (int b = 0; b < B; ++b) {
        const int cur = b & 1;
        if (b + 1 < B) {
            // Issue next batch into the other buffer, then wait until only
            // those 2 remain outstanding => current buffer is complete.
            const float* basen =
                x + ((size_t)(b + 1) * (size_t)T_in + (size_t)idx) * D_DIM;
            async_ld16(ldsL[cur ^ 1], basen + lane4);
            async_ld16(ldsR[cur ^ 1], basen + D_DIM + lane4);
            wait_async_le2();
        } else {
            wait_async_le0();
        }

        // Each lane reads back exactly the bytes it staged: no barrier needed.
        v4f l = *(const v4f*)(&sbuf[cur][lane4]);
        v4f r = *(const v4f*)(&sbuf[cur][D_DIM + lane4]);

        v4f o;
        o.x = __builtin_fmaf(w, r.x - l.x, l.x);
        o.y = __builtin_fmaf(w, r.y - l.y, l.y);
        o.z = __builtin_fmaf(w, r.z - l.z, l.z);
        o.w = __builtin_fmaf(w, r.w - l.w, l.w);

        // Non-temporal store: stream output, keep x in L2.
        v4f* dst = (v4f*)(out + ((size_t)b * (size_t)out_len + (size_t)t) * D_DIM
                              + (size_t)lane4);
        __builtin_nontemporal_store(o, dst);
    }
}

extern "C" void kernel_launch(void* const* d_in, const int* in_sizes, int n_in,
                              void* d_out, int out_size, void* d_ws, size_t ws_size,
                              hipStream_t stream)
{
    const float* x   = (const float*)d_in[0];
    const int*   tl  = (const int*)  d_in[1];
    float*       out = (float*)d_out;

    const int       T_in = in_sizes[1];                   // 4096
    const long long BD   = (long long)in_sizes[0] / T_in; // B*D
    const int       B    = (int)(BD / D_DIM);             // 8
    const int       out_len = (int)((long long)out_size / BD); // 8191

    dim3 grid((unsigned)out_len);
    interp_async_kernel<<<grid, THREADS, 0, stream>>>(x, tl, out, T_in, out_len, B);
}